// SAMBA_19301583028596
// MI455X (gfx1250) — compile-verified
//
#include <hip/hip_runtime.h>
#include <math.h>

typedef float v2f __attribute__((ext_vector_type(2)));
typedef float v8f __attribute__((ext_vector_type(8)));

#define Bc    4
#define Lc    512
#define DMc   256
#define DIc   512
#define Nst   16
#define HIDc  512
#define Mrows (Bc*Lc)   // 2048

__device__ __forceinline__ float sigm_(float x){ return 1.0f/(1.0f+__expf(-x)); }

__device__ __forceinline__ float blockReduce256(float v, float* red){
  int t = threadIdx.x;
  red[t] = v; __syncthreads();
  #pragma unroll
  for (int s = 128; s > 0; s >>= 1){ if (t < s) red[t] += red[t+s]; __syncthreads(); }
  float r = red[0]; __syncthreads();
  return r;
}

// ---------------------------------------------------------------------------
// WMMA fp32 GEMM:  C[M,N] = act( A[M,K](lda) * W[N,K]^T(ldw) + bias[N] )
// One wave computes a 16 x (16*NT) strip, reusing the A fragment NT times.
// Inner loop processes K in chunks of 16: all (NT+1)*4 b64 loads are issued
// before the 4*NT WMMAs so many loads stay in flight (avoids the
// load -> s_wait_loadcnt 0 -> wmma serialization seen previously).
// ACT: 0=none, 1=relu, 2=softplus
// ---------------------------------------------------------------------------
template<int NT, int ACT>
__global__ void gemm_wmma_f32(const float* __restrict__ A, int lda,
                              const float* __restrict__ W, int ldw,
                              const float* __restrict__ bias,
                              float* __restrict__ C, int ldc,
                              int M, int N, int K)
{
  const int lane = threadIdx.x & 31;
  const int wave = threadIdx.x >> 5;
  const int tile = blockIdx.x * (blockDim.x >> 5) + wave;
  const int nStrips = N / (16*NT);
  const int mTiles  = M >> 4;
  if (tile >= mTiles * nStrips) return;   // wave-uniform: EXEC stays all-1s for WMMA
  const int mt = tile / nStrips;
  const int ns = tile - mt * nStrips;

  const int r  = lane & 15;   // row within tile (A) / out column within tile (W, C)
  const int hi = lane >> 4;   // K-half select per ISA 16x4 f32 layout

  const float* Arow = A + (size_t)(mt*16 + r)*lda + 2*hi;
  const float* Wrow[NT];
  #pragma unroll
  for (int j = 0; j < NT; ++j)
    Wrow[j] = W + (size_t)((ns*NT + j)*16 + r)*ldw + 2*hi;

  v8f acc[NT];
  #pragma unroll
  for (int j = 0; j < NT; ++j){
    #pragma unroll
    for (int e = 0; e < 8; ++e) acc[j][e] = 0.0f;
  }

  // K in chunks of 16 (4 WMMA k-steps). All K used here are multiples of 16.
  #pragma clang loop unroll(disable)
  for (int kk = 0; kk < K; kk += 16){
    v2f a[4];
    v2f bf[NT][4];
    #pragma unroll
    for (int u = 0; u < 4; ++u)
      a[u] = *(const v2f*)(Arow + kk + 4*u);
    #pragma unroll
    for (int j = 0; j < NT; ++j){
      #pragma unroll
      for (int u = 0; u < 4; ++u)
        bf[j][u] = *(const v2f*)(Wrow[j] + kk + 4*u);
    }
    #pragma unroll
    for (int u = 0; u < 4; ++u){
      #pragma unroll
      for (int j = 0; j < NT; ++j){
        acc[j] = __builtin_amdgcn_wmma_f32_16x16x4_f32(false, a[u], false, bf[j][u],
                                                       (short)0, acc[j], false, false);
      }
    }
  }

  const int mBase = mt*16 + 8*hi;   // C/D layout: vgpr rr -> M = rr + 8*hi, N = lane&15
  #pragma unroll
  for (int j = 0; j < NT; ++j){
    const int nOut = (ns*NT + j)*16 + r;
    const float bv = bias ? bias[nOut] : 0.0f;
    #pragma unroll
    for (int rr = 0; rr < 8; ++rr){
      float v = acc[j][rr] + bv;
      if (ACT == 1) v = fmaxf(v, 0.0f);
      if (ACT == 2) v = (v > 20.0f) ? v : log1pf(__expf(v));
      C[(size_t)(mBase + rr)*ldc + nOut] = v;
    }
  }
}

// ---------------------------------------------------------------------------
// LayerNorm over DM (=256) with optional sequence reversal on read.
// grid = B*L blocks, 256 threads.
// ---------------------------------------------------------------------------
__global__ void ln_dm_kernel(const float* __restrict__ x,
                             const float* __restrict__ g,
                             const float* __restrict__ bta,
                             float* __restrict__ out, int rev)
{
  __shared__ float red[256];
  const int row = blockIdx.x;
  const int b = row / Lc, l = row % Lc;
  const int lsrc = rev ? (Lc-1-l) : l;
  const int t = threadIdx.x;
  const float v = x[((size_t)b*Lc + lsrc)*DMc + t];
  const float sum  = blockReduce256(v,   red);
  const float sumq = blockReduce256(v*v, red);
  const float mu  = sum * (1.0f/DMc);
  const float var = sumq * (1.0f/DMc) - mu*mu;
  const float rs  = rsqrtf(var + 1e-5f);
  out[(size_t)row*DMc + t] = (v - mu)*rs*g[t] + bta[t];
}

// final: out = LN(x + residual) * g + b, over DM
__global__ void final_ln_kernel(const float* __restrict__ x,
                                const float* __restrict__ resid,
                                const float* __restrict__ g,
                                const float* __restrict__ bta,
                                float* __restrict__ out)
{
  __shared__ float red[256];
  const int row = blockIdx.x;
  const int t = threadIdx.x;
  const float v = x[(size_t)row*DMc + t] + resid[(size_t)row*DMc + t];
  const float sum  = blockReduce256(v,   red);
  const float sumq = blockReduce256(v*v, red);
  const float mu  = sum * (1.0f/DMc);
  const float var = sumq * (1.0f/DMc) - mu*mu;
  const float rs  = rsqrtf(var + 1e-5f);
  out[(size_t)row*DMc + t] = (v - mu)*rs*g[t] + bta[t];
}

// ---------------------------------------------------------------------------
// Depthwise causal conv (DC=4) + SiLU.  xr is (B*L, 1024); channels 0..511.
// ---------------------------------------------------------------------------
__global__ void conv_silu_kernel(const float* __restrict__ xr,
                                 const float* __restrict__ cw,
                                 const float* __restrict__ cb,
                                 float* __restrict__ xc)
{
  const int idx = blockIdx.x*blockDim.x + threadIdx.x;
  if (idx >= Bc*Lc*DIc) return;
  const int ch = idx % DIc;
  const int l  = (idx / DIc) % Lc;
  const int b  = idx / (DIc*Lc);
  const float* w = cw + (size_t)ch*4;
  float acc = cb[ch];
  #pragma unroll
  for (int j = 0; j < 4; ++j){
    const int ls = l - 3 + j;
    if (ls >= 0) acc += w[j] * xr[((size_t)b*Lc + ls)*(2*DIc) + ch];
  }
  xc[idx] = acc * sigm_(acc);
}

// ---------------------------------------------------------------------------
// Selective scan, fused with gating by SiLU(res).
// grid = B blocks, DI (=512) threads. Each thread owns 16-entry state in regs.
// xdbl rows: [0:16]=dt raw, [16:32]=Bm, [32:48]=Cm.
// ---------------------------------------------------------------------------
__global__ void scan_kernel(const float* __restrict__ xdbl,
                            const float* __restrict__ dlt,
                            const float* __restrict__ xc,
                            const float* __restrict__ xr,
                            const float* __restrict__ A_log,
                            const float* __restrict__ Dp,
                            float* __restrict__ y)
{
  __shared__ float sBC[32];      // [0:16]=Bm row, [16:32]=Cm row
  const int b = blockIdx.x;
  const int d = threadIdx.x;
  float Areg[Nst], s[Nst];
  #pragma unroll
  for (int n = 0; n < Nst; ++n){
    Areg[n] = -__expf(A_log[(size_t)d*Nst + n]);
    s[n] = 0.0f;
  }
  const float Dv = Dp[d];
  for (int t = 0; t < Lc; ++t){
    const size_t row = (size_t)b*Lc + t;
    if (threadIdx.x < 32) sBC[threadIdx.x] = xdbl[row*48 + 16 + threadIdx.x];
    __syncthreads();
    const float dv = dlt[row*DIc + d];
    const float u  = xc[row*DIc + d];
    const float du = dv * u;
    float yv = 0.0f;
    #pragma unroll
    for (int n = 0; n < Nst; ++n){
      const float dA = __expf(dv * Areg[n]);
      s[n] = dA * s[n] + du * sBC[n];
      yv += s[n] * sBC[16 + n];
    }
    yv += u * Dv;
    const float res = xr[row*(2*DIc) + DIc + d];
    yv *= res * sigm_(res);
    y[row*DIc + d] = yv;
    __syncthreads();
  }
}

// x = m0 + reverse_L(m1) + 3*x
__global__ void combine_kernel(float* __restrict__ x,
                               const float* __restrict__ m0,
                               const float* __restrict__ m1)
{
  const int idx = blockIdx.x*blockDim.x + threadIdx.x;
  if (idx >= Bc*Lc*DMc) return;
  const int c = idx % DMc;
  const int l = (idx / DMc) % Lc;
  const int b = idx / (DMc*Lc);
  const float mrev = m1[((size_t)b*Lc + (Lc-1-l))*DMc + c];
  x[idx] = m0[idx] + mrev + 3.0f*x[idx];
}

// Norm over L (=512) of x^T: rows are (b,dm). mode 0 = LN(w,b), 1 = RMS(w).
// grid = B*DM blocks, 256 threads, 2 elements per thread. Output ht (B*DM, L).
__global__ void norm_L_kernel(const float* __restrict__ x,
                              const float* __restrict__ w,
                              const float* __restrict__ bta,
                              float* __restrict__ ht, int mode)
{
  __shared__ float red[256];
  const int row = blockIdx.x;
  const int b = row / DMc, dm = row % DMc;
  const int t = threadIdx.x;
  const float v0 = x[((size_t)b*Lc + t      )*DMc + dm];
  const float v1 = x[((size_t)b*Lc + t + 256)*DMc + dm];
  const float sum  = blockReduce256(v0 + v1,       red);
  const float sumq = blockReduce256(v0*v0 + v1*v1, red);
  float o0, o1;
  if (mode == 0){
    const float mu  = sum * (1.0f/Lc);
    const float var = sumq * (1.0f/Lc) - mu*mu;
    const float rs  = rsqrtf(var + 1e-5f);
    o0 = (v0 - mu)*rs*w[t]       + bta[t];
    o1 = (v1 - mu)*rs*w[t + 256] + bta[t + 256];
  } else {
    const float rs = rsqrtf(sumq * (1.0f/Lc) + 1e-5f);
    o0 = v0*rs*w[t];
    o1 = v1*rs*w[t + 256];
  }
  ht[(size_t)row*Lc + t      ] = o0;
  ht[(size_t)row*Lc + t + 256] = o1;
}

// x[b,l,dm] += h2[(b*DM+dm)*L + l]
__global__ void add_transpose_kernel(float* __restrict__ x,
                                     const float* __restrict__ h2)
{
  const int idx = blockIdx.x*blockDim.x + threadIdx.x;
  if (idx >= Bc*Lc*DMc) return;
  const int c = idx % DMc;
  const int l = (idx / DMc) % Lc;
  const int b = idx / (DMc*Lc);
  x[idx] += h2[((size_t)b*DMc + c)*Lc + l];
}

// ---------------------------------------------------------------------------
extern "C" void kernel_launch(void* const* d_in, const int* in_sizes, int n_in,
                              void* d_out, int out_size, void* d_ws, size_t ws_size,
                              hipStream_t stream)
{
  const float* input_ids  = (const float*)d_in[0];
  const float* emb_W      = (const float*)d_in[1];
  const float* emb_b      = (const float*)d_in[2];
  const float* ln_g       = (const float*)d_in[3];
  const float* ln_b       = (const float*)d_in[4];
  const float* inproj_W   = (const float*)d_in[5];
  const float* inproj_b   = (const float*)d_in[6];
  const float* conv_w     = (const float*)d_in[7];
  const float* conv_b     = (const float*)d_in[8];
  const float* xproj_W    = (const float*)d_in[9];
  const float* dtproj_W   = (const float*)d_in[10];
  const float* dtproj_b   = (const float*)d_in[11];
  const float* outproj_W  = (const float*)d_in[12];
  const float* outproj_b  = (const float*)d_in[13];
  const float* A_log      = (const float*)d_in[14];
  const float* Dp         = (const float*)d_in[15];
  const float* lin_norm_w = (const float*)d_in[16];
  const float* lin_norm_b = (const float*)d_in[17];
  const float* lin_W1     = (const float*)d_in[18];
  const float* lin_b1     = (const float*)d_in[19];
  const float* lin_W2     = (const float*)d_in[20];
  const float* lin_b2     = (const float*)d_in[21];
  const float* normf_g    = (const float*)d_in[22];
  const float* normf_b    = (const float*)d_in[23];

  float* ws = (float*)d_ws;
  size_t off = 0;
  auto take = [&](size_t n){ float* p = ws + off; off += (n + 255) & ~(size_t)255; return p; };
  float* residual = take((size_t)Mrows*DMc);
  float* x    = take((size_t)Mrows*DMc);
  float* xn   = take((size_t)Mrows*DMc);
  float* xr   = take((size_t)Mrows*2*DIc);
  float* xc   = take((size_t)Mrows*DIc);
  float* xdbl = take((size_t)Mrows*48);
  float* dlt  = take((size_t)Mrows*DIc);
  float* yb   = take((size_t)Mrows*DIc);
  float* m0   = take((size_t)Mrows*DMc);
  float* m1   = take((size_t)Mrows*DMc);
  float* ht   = take((size_t)Bc*DMc*Lc);
  float* h1   = take((size_t)Bc*DMc*HIDc);
  float* h2   = take((size_t)Bc*DMc*Lc);
  (void)ws_size; (void)in_sizes; (void)n_in; (void)out_size;

  auto nblk = [](int tiles){ return (tiles + 7) / 8; };   // 8 waves / block

  // ---- embedding: x = ids @ emb_W^T + emb_b ; residual = x
  {
    const int tiles = (Mrows/16) * (DMc/64);
    gemm_wmma_f32<4,0><<<nblk(tiles), 256, 0, stream>>>(
        input_ids, 256, emb_W, 256, emb_b, x, DMc, Mrows, DMc, 256);
    hipMemcpyAsync(residual, x, sizeof(float)*(size_t)Mrows*DMc,
                   hipMemcpyDeviceToDevice, stream);
  }

  for (int i = 0; i < 4; ++i){
    for (int d = 0; d < 2; ++d){
      const int li = d*4 + i;
      const float* g    = ln_g      + (size_t)li*DMc;
      const float* bt   = ln_b      + (size_t)li*DMc;
      const float* inW  = inproj_W  + (size_t)li*2*DIc*DMc;
      const float* inB  = inproj_b  + (size_t)li*2*DIc;
      const float* cw   = conv_w    + (size_t)li*DIc*4;
      const float* cb   = conv_b    + (size_t)li*DIc;
      const float* xW   = xproj_W   + (size_t)li*48*DIc;
      const float* dtW  = dtproj_W  + (size_t)li*DIc*16;
      const float* dtB  = dtproj_b  + (size_t)li*DIc;
      const float* oW   = outproj_W + (size_t)li*DMc*DIc;
      const float* oB   = outproj_b + (size_t)li*DMc;
      const float* Al   = A_log     + (size_t)li*DIc*Nst;
      const float* Dpl  = Dp        + (size_t)li*DIc;

      // LN over DM (reversed read for backward direction)
      ln_dm_kernel<<<Mrows, 256, 0, stream>>>(x, g, bt, xn, d);

      // inproj: (2048,256) @ (256,1024)^T
      gemm_wmma_f32<4,0><<<nblk((Mrows/16)*(1024/64)), 256, 0, stream>>>(
          xn, DMc, inW, DMc, inB, xr, 2*DIc, Mrows, 2*DIc, DMc);

      // depthwise causal conv + SiLU
      conv_silu_kernel<<<(Mrows*DIc + 255)/256, 256, 0, stream>>>(xr, cw, cb, xc);

      // xproj: (2048,512) @ (512,48)^T  (NT=3 strip covers N=48)
      gemm_wmma_f32<3,0><<<nblk((Mrows/16)*1), 256, 0, stream>>>(
          xc, DIc, xW, DIc, nullptr, xdbl, 48, Mrows, 48, DIc);

      // dtproj + softplus: (2048,16 view of xdbl) @ (16,512)^T
      gemm_wmma_f32<4,2><<<nblk((Mrows/16)*(DIc/64)), 256, 0, stream>>>(
          xdbl, 48, dtW, 16, dtB, dlt, DIc, Mrows, DIc, 16);

      // selective scan + gating
      scan_kernel<<<Bc, DIc, 0, stream>>>(xdbl, dlt, xc, xr, Al, Dpl, yb);

      // outproj: (2048,512) @ (512,256)^T
      gemm_wmma_f32<4,0><<<nblk((Mrows/16)*(DMc/64)), 256, 0, stream>>>(
          yb, DIc, oW, DIc, oB, (d == 0 ? m0 : m1), DMc, Mrows, DMc, DIc);
    }

    // x = m_fwd + reverse(m_bwd) + 3x
    combine_kernel<<<(Mrows*DMc + 255)/256, 256, 0, stream>>>(x, m0, m1);

    // channel-mixing MLP on x^T
    norm_L_kernel<<<Bc*DMc, 256, 0, stream>>>(
        x, lin_norm_w + (size_t)i*Lc, lin_norm_b + (size_t)i*Lc, ht, (i == 0) ? 0 : 1);

    gemm_wmma_f32<4,1><<<nblk((Bc*DMc/16)*(HIDc/64)), 256, 0, stream>>>(
        ht, Lc, lin_W1 + (size_t)i*HIDc*Lc, Lc, lin_b1 + (size_t)i*HIDc,
        h1, HIDc, Bc*DMc, HIDc, Lc);

    gemm_wmma_f32<4,0><<<nblk((Bc*DMc/16)*(Lc/64)), 256, 0, stream>>>(
        h1, HIDc, lin_W2 + (size_t)i*Lc*HIDc, HIDc, lin_b2 + (size_t)i*Lc,
        h2, Lc, Bc*DMc, Lc, HIDc);

    add_transpose_kernel<<<(Mrows*DMc + 255)/256, 256, 0, stream>>>(x, h2);
  }

  // final layer norm of (x + residual)
  final_ln_kernel<<<Mrows, 256, 0, stream>>>(x, residual, normf_g, normf_b,
                                             (float*)d_out);
}